// SpatialAttention_55980603736222
// MI455X (gfx1250) — compile-verified
//
#include <hip/hip_runtime.h>

// Problem geometry (fixed by the reference).
#define HW_          50176          // 224*224
#define CHN_         128
#define BATCH_       32
#define PIX4_PER_IMG (HW_ / 4)      // 12544 float4 groups per image
#define TOTAL4_      (BATCH_ * PIX4_PER_IMG)  // 401408
#define BLK_         256            // 8 wave32s per block
#define DEPTH_       8              // async pipeline depth (ASYNCcnt max is 63)

typedef __attribute__((ext_vector_type(4))) float f32x4;  // native vector for NT store

// ---------------------------------------------------------------------------
// Main streaming kernel: channel max+sum reduction via an 8-deep
// global->LDS async pipeline (CDNA5 ASYNCcnt path), then fused affine+sigmoid.
// Each thread owns 4 consecutive pixels (one float4 per channel step).
// Roofline: 822 MB read / 23.3 TB/s => ~35 us floor; purely bandwidth bound.
// ---------------------------------------------------------------------------
__global__ __launch_bounds__(BLK_) void sa_main_kernel(const float* __restrict__ x,
                                                       const float* __restrict__ coef,
                                                       float* __restrict__ out) {
    __shared__ float4 stage[DEPTH_][BLK_];   // 32 KB: per-lane private slots, no barrier needed

    const int tid = threadIdx.x;
    const int g   = blockIdx.x * BLK_ + tid;   // float4-group index, grid sized exactly
    const int img = g / PIX4_PER_IMG;
    const int r4  = g - img * PIX4_PER_IMG;
    const float* gp = x + (size_t)img * (size_t)(CHN_ * HW_) + (size_t)r4 * 4u;

    float4 mx, sm;
    mx.x = mx.y = mx.z = mx.w = -__builtin_inff();
    sm.x = sm.y = sm.z = sm.w = 0.0f;

#if defined(__gfx1250__)
    // 32-bit LDS byte address of this lane's slot 0 (low 32 bits of the flat
    // shared-aperture address are the wave-relative LDS offset).
    const unsigned lds0 = (unsigned)(unsigned long long)(&stage[0][tid]);

    // Prime the pipeline: channels 0..DEPTH-1 in flight.
    #pragma unroll
    for (int s = 0; s < DEPTH_; ++s) {
        unsigned la = lds0 + (unsigned)s * (BLK_ * 16u);
        unsigned long long ga = (unsigned long long)(gp + (size_t)s * HW_);
        asm volatile("global_load_async_to_lds_b128 %0, %1, off"
                     :: "v"(la), "v"(ga) : "memory");
    }

    // Steady state: consume channel c from LDS, refill its slot with channel
    // c+DEPTH. Unroll by the pipeline depth so the slot index (c & 7) is a
    // compile-time constant in every unrolled instance.
    #pragma unroll 8
    for (int c = 0; c < CHN_ - DEPTH_; ++c) {
        asm volatile("s_wait_asynccnt 7" ::: "memory");   // oldest of 8 has landed
        float4 v = stage[c & (DEPTH_ - 1)][tid];
        mx.x = fmaxf(mx.x, v.x); mx.y = fmaxf(mx.y, v.y);
        mx.z = fmaxf(mx.z, v.z); mx.w = fmaxf(mx.w, v.w);
        sm.x += v.x; sm.y += v.y; sm.z += v.z; sm.w += v.w;
        unsigned la = lds0 + (unsigned)(c & (DEPTH_ - 1)) * (BLK_ * 16u);
        unsigned long long ga = (unsigned long long)(gp + (size_t)(c + DEPTH_) * HW_);
        // v.x as dummy operand: guarantees the ds_load consumed the slot before reissue.
        asm volatile("global_load_async_to_lds_b128 %0, %1, off"
                     :: "v"(la), "v"(ga), "v"(v.x) : "memory");
    }

    // Drain: last DEPTH channels already in their slots.
    asm volatile("s_wait_asynccnt 0" ::: "memory");
    #pragma unroll
    for (int c = CHN_ - DEPTH_; c < CHN_; ++c) {
        float4 v = stage[c & (DEPTH_ - 1)][tid];
        mx.x = fmaxf(mx.x, v.x); mx.y = fmaxf(mx.y, v.y);
        mx.z = fmaxf(mx.z, v.z); mx.w = fmaxf(mx.w, v.w);
        sm.x += v.x; sm.y += v.y; sm.z += v.z; sm.w += v.w;
    }
#else
    // Portable fallback (host pass / non-gfx1250): direct coalesced float4 loads.
    for (int c = 0; c < CHN_; ++c) {
        float4 v = *(const float4*)(gp + (size_t)c * HW_);
        mx.x = fmaxf(mx.x, v.x); mx.y = fmaxf(mx.y, v.y);
        mx.z = fmaxf(mx.z, v.z); mx.w = fmaxf(mx.w, v.w);
        sm.x += v.x; sm.y += v.y; sm.z += v.z; sm.w += v.w;
    }
#endif

    // Collapsed conv chain + sigmoid. coef[] is uniform -> scalar loads.
    const float A = coef[0], Bv = coef[1], Cv = coef[2];
    float tx = fmaf(A, mx.x, fmaf(Bv, sm.x, Cv));
    float ty = fmaf(A, mx.y, fmaf(Bv, sm.y, Cv));
    float tz = fmaf(A, mx.z, fmaf(Bv, sm.z, Cv));
    float tw = fmaf(A, mx.w, fmaf(Bv, sm.w, Cv));
    f32x4 r;
    r.x = 1.0f / (1.0f + __expf(-tx));
    r.y = 1.0f / (1.0f + __expf(-ty));
    r.z = 1.0f / (1.0f + __expf(-tz));
    r.w = 1.0f / (1.0f + __expf(-tw));

    // Output is never re-read by this dispatch: keep it out of GL2 (th:NT).
    __builtin_nontemporal_store(r, (f32x4*)(out + (size_t)g * 4u));
}

// ---------------------------------------------------------------------------
// Prologue: collapse the 4 linear 1x1 convs into out = sigmoid(A*max + B*avg + C).
// Runs as a single thread; the compiler lowers it to pure SALU float math.
// ---------------------------------------------------------------------------
__global__ void sa_compose_kernel(const float* __restrict__ w1, const float* __restrict__ b1,
                                  const float* __restrict__ w2, const float* __restrict__ b2,
                                  const float* __restrict__ w3, const float* __restrict__ b3,
                                  const float* __restrict__ w4, const float* __restrict__ b4,
                                  float* __restrict__ coef) {
    if (threadIdx.x != 0 || blockIdx.x != 0) return;

    // Layer 1: [8,2] applied to (max, avg) + b1
    float a1[8], c1[8], v1[8];
    #pragma unroll
    for (int o = 0; o < 8; ++o) { a1[o] = w1[o*2+0]; c1[o] = w1[o*2+1]; v1[o] = b1[o]; }
    // Layer 2: [4,8]
    float a2[4], c2[4], v2[4];
    #pragma unroll
    for (int o = 0; o < 4; ++o) {
        float a = 0.f, c = 0.f, v = b2[o];
        #pragma unroll
        for (int k = 0; k < 8; ++k) { float w = w2[o*8+k]; a += w*a1[k]; c += w*c1[k]; v += w*v1[k]; }
        a2[o] = a; c2[o] = c; v2[o] = v;
    }
    // Layer 3: [2,4]
    float a3[2], c3[2], v3[2];
    #pragma unroll
    for (int o = 0; o < 2; ++o) {
        float a = 0.f, c = 0.f, v = b3[o];
        #pragma unroll
        for (int k = 0; k < 4; ++k) { float w = w3[o*4+k]; a += w*a2[k]; c += w*c2[k]; v += w*v2[k]; }
        a3[o] = a; c3[o] = c; v3[o] = v;
    }
    // Layer 4: [1,2]
    float A = 0.f, B = 0.f, C = b4[0];
    #pragma unroll
    for (int k = 0; k < 2; ++k) { float w = w4[k]; A += w*a3[k]; B += w*c3[k]; C += w*v3[k]; }

    coef[0] = A;
    coef[1] = B * (1.0f / (float)CHN_);  // fold the mean's 1/128 into the avg coefficient
    coef[2] = C;
}

// ---------------------------------------------------------------------------
// Harness entry point.
// Inputs (setup_inputs order): x, w1, b1, w2, b2, w3, b3, w4, b4 — all float32.
// Output: [32,1,224,224] float32 = 1,605,632 elements.
// ---------------------------------------------------------------------------
extern "C" void kernel_launch(void* const* d_in, const int* in_sizes, int n_in,
                              void* d_out, int out_size, void* d_ws, size_t ws_size,
                              hipStream_t stream) {
    const float* x  = (const float*)d_in[0];
    const float* w1 = (const float*)d_in[1];
    const float* b1 = (const float*)d_in[2];
    const float* w2 = (const float*)d_in[3];
    const float* b2 = (const float*)d_in[4];
    const float* w3 = (const float*)d_in[5];
    const float* b3 = (const float*)d_in[6];
    const float* w4 = (const float*)d_in[7];
    const float* b4 = (const float*)d_in[8];
    float* coef = (float*)d_ws;          // 3 floats of scratch
    float* out  = (float*)d_out;

    sa_compose_kernel<<<1, 32, 0, stream>>>(w1, b1, w2, b2, w3, b3, w4, b4, coef);

    const int nblocks = TOTAL4_ / BLK_;  // 401408 / 256 = 1568, exact
    sa_main_kernel<<<nblocks, BLK_, 0, stream>>>(x, coef, out);
}